// GeneTreeGIN_65867618451931
// MI455X (gfx1250) — compile-verified
//
#include <hip/hip_runtime.h>
#include <hip/hip_bf16.h>

typedef __attribute__((ext_vector_type(2))) float v2f;
typedef __attribute__((ext_vector_type(8))) float v8f;

__device__ __forceinline__ v8f wmma_f32(v2f a, v2f b, v8f c) {
  // D = A(16x4 f32) * B(4x16 f32) + C(16x16 f32)
  return __builtin_amdgcn_wmma_f32_16x16x4_f32(false, a, false, b, (short)0, c,
                                               false, false);
}

// ---------------------------------------------------------------------------
// x[node,d] = emb_table[sp<0 ? S : sp, d]
__global__ void k_embed(const int* __restrict__ sp, const float* __restrict__ emb,
                        float* __restrict__ x, int ntot, int S, int D) {
  int id = blockIdx.x * blockDim.x + threadIdx.x;
  if (id >= ntot * D) return;
  int node = id / D, d = id - node * D;
  int s = sp[node];
  int row = (s < 0) ? S : s;
  x[id] = emb[(size_t)row * D + d];
}

// h = (1+eps[l]) * x
__global__ void k_init_h(const float* __restrict__ x, float* __restrict__ h,
                         const float* __restrict__ eps, int l, int n) {
  int id = blockIdx.x * blockDim.x + threadIdx.x;
  if (id >= n) return;
  h[id] = (1.0f + eps[l]) * x[id];
}

// h[dst,:] += x[src,:]  (per directed edge, 4 floats per thread)
__global__ void k_scatter(const int* __restrict__ ei, const float* __restrict__ x,
                          float* __restrict__ h, int nDir, int D) {
  int chunks = D >> 2;
  int id = blockIdx.x * blockDim.x + threadIdx.x;
  if (id >= nDir * chunks) return;
  int e = id / chunks;
  int c = (id - e * chunks) << 2;
  int src = ei[e];
  int dst = ei[nDir + e];
  const float4 v = *(const float4*)(x + (size_t)src * D + c);
  float* hd = h + (size_t)dst * D + c;
  atomicAdd(hd + 0, v.x);
  atomicAdd(hd + 1, v.y);
  atomicAdd(hd + 2, v.z);
  atomicAdd(hd + 3, v.w);
}

// ---------------------------------------------------------------------------
// Fused GIN MLP + residual + LayerNorm, D == 64 only.
// x <- LN(x + relu(h@W1+b1)@W2 + b2)   (in-place on x; h is the aggregated input)
__global__ __launch_bounds__(256) void k_mlp_ln(
    const float* __restrict__ h, float* __restrict__ x,
    const float* __restrict__ W1, const float* __restrict__ b1,
    const float* __restrict__ W2, const float* __restrict__ b2,
    const float* __restrict__ lng, const float* __restrict__ lnb, int ntot) {
  __shared__ v2f sW1[32 * 64];  // pair p, col n: (W1[2p][n], W1[2p+1][n])
  __shared__ v2f sW2[32 * 64];
  __shared__ float sB1[64], sB2[64], sG[64], sBt[64];
  __shared__ __align__(16) float sH[8][16 * 68];  // per-wave 16x64 tile, stride 68

  int tid = threadIdx.x;
  for (int i = tid; i < 32 * 64; i += 256) {
    int p = i >> 6, n = i & 63;
    v2f w1, w2;
    w1[0] = W1[(size_t)(2 * p) * 64 + n];
    w1[1] = W1[(size_t)(2 * p + 1) * 64 + n];
    w2[0] = W2[(size_t)(2 * p) * 64 + n];
    w2[1] = W2[(size_t)(2 * p + 1) * 64 + n];
    sW1[i] = w1;
    sW2[i] = w2;
  }
  if (tid < 64) { sB1[tid] = b1[tid]; sB2[tid] = b2[tid]; sG[tid] = lng[tid]; sBt[tid] = lnb[tid]; }
  __syncthreads();

  int wave = tid >> 5, lane = tid & 31;
  int lhalf = lane >> 4, l16 = lane & 15;
  int row0 = (blockIdx.x * 8 + wave) * 16;
  int r = row0 + l16;  // A-fragment / epilogue row for this lane
  // Clamped row for unconditional A loads: out-of-range rows produce garbage
  // only in C rows that are never stored (epilogue is guarded by r < ntot).
  int rc = (r < ntot) ? r : (ntot - 1);
  const float* hrow = h + (size_t)rc * 64 + lhalf * 2;
  float* myH = &sH[wave][0];

  v8f acc0, acc1, acc2, acc3;
  {
    float c0 = sB1[l16], c1 = sB1[16 + l16], c2 = sB1[32 + l16], c3 = sB1[48 + l16];
#pragma unroll
    for (int j = 0; j < 8; ++j) { acc0[j] = c0; acc1[j] = c1; acc2[j] = c2; acc3[j] = c3; }
  }
  // GEMM1: h @ W1  (K = 64 in steps of 4)
#pragma unroll 4
  for (int k = 0; k < 16; ++k) {
    v2f a = *(const v2f*)(hrow + k * 4);
    int bi = (2 * k + lhalf) * 64 + l16;
    acc0 = wmma_f32(a, sW1[bi], acc0);
    acc1 = wmma_f32(a, sW1[bi + 16], acc1);
    acc2 = wmma_f32(a, sW1[bi + 32], acc2);
    acc3 = wmma_f32(a, sW1[bi + 48], acc3);
  }
  // ReLU + stash hidden tile (row = j + 8*lhalf, col = 16t + l16)
#pragma unroll
  for (int j = 0; j < 8; ++j) {
    int rr = (j + lhalf * 8) * 68 + l16;
    myH[rr] = fmaxf(acc0[j], 0.f);
    myH[rr + 16] = fmaxf(acc1[j], 0.f);
    myH[rr + 32] = fmaxf(acc2[j], 0.f);
    myH[rr + 48] = fmaxf(acc3[j], 0.f);
  }
  __syncthreads();
  {
    float c0 = sB2[l16], c1 = sB2[16 + l16], c2 = sB2[32 + l16], c3 = sB2[48 + l16];
#pragma unroll
    for (int j = 0; j < 8; ++j) { acc0[j] = c0; acc1[j] = c1; acc2[j] = c2; acc3[j] = c3; }
  }
  // GEMM2: relu(h1) @ W2
#pragma unroll 4
  for (int k = 0; k < 16; ++k) {
    v2f a = *(const v2f*)(myH + l16 * 68 + (k * 4 + lhalf * 2));
    int bi = (2 * k + lhalf) * 64 + l16;
    acc0 = wmma_f32(a, sW2[bi], acc0);
    acc1 = wmma_f32(a, sW2[bi + 16], acc1);
    acc2 = wmma_f32(a, sW2[bi + 32], acc2);
    acc3 = wmma_f32(a, sW2[bi + 48], acc3);
  }
  __syncthreads();
#pragma unroll
  for (int j = 0; j < 8; ++j) {
    int rr = (j + lhalf * 8) * 68 + l16;
    myH[rr] = acc0[j];
    myH[rr + 16] = acc1[j];
    myH[rr + 32] = acc2[j];
    myH[rr + 48] = acc3[j];
  }
  __syncthreads();

  // Residual + LayerNorm: lane handles row l16, half-cols [32*lhalf, +32)
  int c0 = lhalf * 32;
  float s = 0.f;
  if (r < ntot) {
    for (int c = 0; c < 32; ++c) {
      int col = c0 + c;
      float v = x[(size_t)r * 64 + col] + myH[l16 * 68 + col];
      myH[l16 * 68 + col] = v;
      s += v;
    }
  }
  float st = s + __shfl_xor(s, 16, 32);
  float mean = st * (1.0f / 64.0f);
  float q = 0.f;
  if (r < ntot) {
    for (int c = 0; c < 32; ++c) {
      float dv = myH[l16 * 68 + c0 + c] - mean;
      q += dv * dv;
    }
  }
  float qt = q + __shfl_xor(q, 16, 32);
  float rstd = rsqrtf(qt * (1.0f / 64.0f) + 1e-5f);
  if (r < ntot) {
    for (int c = 0; c < 32; ++c) {
      int col = c0 + c;
      float v = (myH[l16 * 68 + col] - mean) * rstd * sG[col] + sBt[col];
      x[(size_t)r * 64 + col] = v;
    }
  }
}

// ---------------------------------------------------------------------------
// Species pooling scatter: Psum[s][d*G+g] += x[node,d]; cnt[g*S+s] += 1
__global__ void k_pool_scatter(const float* __restrict__ x, const int* __restrict__ sp,
                               const int* __restrict__ leaf, const int* __restrict__ batch,
                               float* __restrict__ Psum, float* __restrict__ cnt,
                               int ntot, int S, int G, int D) {
  int id = blockIdx.x * blockDim.x + threadIdx.x;
  if (id >= ntot * D) return;
  int node = id / D, d = id - node * D;
  int s = sp[node];
  if (leaf[node] && s >= 0) {
    int g = batch[node];
    atomicAdd(&Psum[(size_t)s * D * G + (size_t)d * G + g], x[id]);
    if (d == 0) atomicAdd(&cnt[(size_t)g * S + s], 1.0f);
  }
}

// P[s][d*G+g] = Psum / max(cnt[g,s],1)   (in place)
__global__ void k_pool_div(float* __restrict__ P, const float* __restrict__ cnt,
                           int n, int G, int GD, int S) {
  int id = blockIdx.x * blockDim.x + threadIdx.x;
  if (id >= n) return;
  int s = id / GD;
  int g = id % G;
  P[id] = P[id] / fmaxf(cnt[(size_t)g * S + s], 1.0f);
}

// ccnt[e,g] = sum_s mf[e,s] * (cnt[g,s] > 0)
__global__ void k_ccnt(const int* __restrict__ clade, const float* __restrict__ cnt,
                       float* __restrict__ ccnt, int nE, int G, int S) {
  int id = blockIdx.x * blockDim.x + threadIdx.x;
  if (id >= nE * G) return;
  int e = id / G, g = id - e * G;
  const int* mrow = clade + (size_t)e * S;
  const float* crow = cnt + (size_t)g * S;
  float c = 0.f;
  for (int s = 0; s < S; ++s)
    c += (mrow[s] && crow[s] > 0.f) ? 1.f : 0.f;
  ccnt[id] = c;
}

// ---------------------------------------------------------------------------
// csum[e][col] = sum_s mf[e,s] * P[s][col], col = d*G+g.  WMMA f32 16x16x4 GEMM.
__global__ __launch_bounds__(256) void k_clade_gemm(
    const int* __restrict__ clade, const float* __restrict__ P,
    float* __restrict__ csum, int nE, int S, int GD) {
  extern __shared__ float dynsm[];
  v2f* sAp = (v2f*)dynsm;  // [S/2][16]: pair kp, row r: (A[r][2kp], A[r][2kp+1])
  int tid = threadIdx.x;
  int e0 = blockIdx.x * 16;
  int npair = S >> 1;
  for (int i = tid; i < npair * 16; i += 256) {
    int kp = i >> 4, row = i & 15;
    int e = e0 + row;
    v2f a;
    a[0] = 0.f; a[1] = 0.f;
    if (e < nE) {
      a[0] = (float)clade[(size_t)e * S + 2 * kp];
      a[1] = (float)clade[(size_t)e * S + 2 * kp + 1];
    }
    sAp[i] = a;
  }
  __syncthreads();

  int wave = tid >> 5, lane = tid & 31;
  int lhalf = lane >> 4, l16 = lane & 15;
  int colb = blockIdx.y * 512 + wave * 64 + l16;
  int c0 = colb, c1 = colb + 16, c2 = colb + 32, c3 = colb + 48;
  int m0 = (c0 < GD) ? c0 : 0, m1 = (c1 < GD) ? c1 : 0;
  int m2 = (c2 < GD) ? c2 : 0, m3 = (c3 < GD) ? c3 : 0;

  v8f acc0, acc1, acc2, acc3;
#pragma unroll
  for (int j = 0; j < 8; ++j) { acc0[j] = 0.f; acc1[j] = 0.f; acc2[j] = 0.f; acc3[j] = 0.f; }

  int kq = S >> 2;
  for (int k = 0; k < kq; ++k) {
    v2f a = sAp[(2 * k + lhalf) * 16 + l16];
    int kk = 4 * k + 2 * lhalf;
    const float* Pr0 = P + (size_t)kk * GD;
    const float* Pr1 = Pr0 + GD;
    v2f b;
    b[0] = Pr0[m0]; b[1] = Pr1[m0]; acc0 = wmma_f32(a, b, acc0);
    b[0] = Pr0[m1]; b[1] = Pr1[m1]; acc1 = wmma_f32(a, b, acc1);
    b[0] = Pr0[m2]; b[1] = Pr1[m2]; acc2 = wmma_f32(a, b, acc2);
    b[0] = Pr0[m3]; b[1] = Pr1[m3]; acc3 = wmma_f32(a, b, acc3);
  }
#pragma unroll
  for (int j = 0; j < 8; ++j) {
    int e = e0 + j + lhalf * 8;
    if (e < nE) {
      float* C = csum + (size_t)e * GD;
      if (c0 < GD) C[c0] = acc0[j];
      if (c1 < GD) C[c1] = acc1[j];
      if (c2 < GD) C[c2] = acc2[j];
      if (c3 < GD) C[c3] = acc3[j];
    }
  }
}

// ---------------------------------------------------------------------------
// Per edge e: mean/std over gene trees of gt_means = csum/max(ccnt,1)
__global__ void k_finalize(const float* __restrict__ csum, const float* __restrict__ ccnt,
                           float* __restrict__ out, int G, int D) {
  extern __shared__ float dynsm[];
  float* sInv = dynsm;            // [G]
  float* sHas = dynsm + G;        // [G]
  float* sPart = dynsm + 2 * G;   // [blockDim]
  int e = blockIdx.x;
  int tid = threadIdx.x;
  float pc = 0.f;
  for (int g = tid; g < G; g += blockDim.x) {
    float c = ccnt[(size_t)e * G + g];
    float hv = (c > 0.f) ? 1.f : 0.f;
    sHas[g] = hv;
    sInv[g] = 1.0f / fmaxf(c, 1.0f);
    pc += hv;
  }
  sPart[tid] = pc;
  __syncthreads();
  if (tid == 0) {
    float nvv = 0.f;
    for (int i = 0; i < blockDim.x; ++i) nvv += sPart[i];
    sPart[0] = nvv;
  }
  __syncthreads();
  float nv = sPart[0];
  int d = tid;
  if (d < D) {
    const float* row = csum + (size_t)e * D * G + (size_t)d * G;
    float s = 0.f;
    for (int g = 0; g < G; ++g) s += row[g] * sInv[g] * sHas[g];
    float mean = s / fmaxf(nv, 1.0f);
    float q = 0.f;
    for (int g = 0; g < G; ++g) {
      float dv = (row[g] * sInv[g] - mean) * sHas[g];
      q += dv * dv;
    }
    float var = q / fmaxf(nv - 1.0f, 1.0f);
    float stdv = (nv > 1.0f) ? sqrtf(var) : 0.f;
    float meano = (nv > 0.f) ? mean : 0.f;
    out[(size_t)e * 2 * D + d] = meano;
    out[(size_t)e * 2 * D + D + d] = stdv;
  }
}

// ---------------------------------------------------------------------------
extern "C" void kernel_launch(void* const* d_in, const int* in_sizes, int n_in,
                              void* d_out, int out_size, void* d_ws, size_t ws_size,
                              hipStream_t stream) {
  const int* ei = (const int*)d_in[0];
  const int* sp = (const int*)d_in[1];
  const int* leaf = (const int*)d_in[2];
  const int* batch = (const int*)d_in[3];
  const int* clade = (const int*)d_in[4];
  // d_in[5] = n_gt (derived host-side from sizes instead)
  const float* emb = (const float*)d_in[6];
  const float* W1 = (const float*)d_in[7];
  const float* b1 = (const float*)d_in[8];
  const float* W2 = (const float*)d_in[9];
  const float* b2 = (const float*)d_in[10];
  const float* eps = (const float*)d_in[11];
  const float* lng = (const float*)d_in[12];
  const float* lnb = (const float*)d_in[13];
  float* out = (float*)d_out;

  const int ntot = in_sizes[1];
  const int L = in_sizes[11];
  const int D = in_sizes[8] / L;          // b1 is [L, D]
  const int S = in_sizes[6] / D - 1;      // emb is [S+1, D]
  const int nDir = in_sizes[0] / 2;       // directed edges
  const int G = ntot - in_sizes[0] / 4;   // trees: undirected edges = ntot - G
  const int nE = in_sizes[4] / S;         // species-tree edges
  const int GD = G * D;

  char* w = (char*)d_ws;
  auto carve = [&](size_t bytes) -> char* {
    char* p = w;
    w += (bytes + 255) & ~(size_t)255;
    return p;
  };
  float* x = (float*)carve((size_t)ntot * D * 4);
  float* h = (float*)carve((size_t)ntot * D * 4);
  float* P = (float*)carve((size_t)S * GD * 4);
  float* cnt = (float*)carve((size_t)G * S * 4);
  float* csum = (float*)carve((size_t)nE * GD * 4);
  float* ccn = (float*)carve((size_t)nE * G * 4);

  hipMemsetAsync(P, 0, (size_t)S * GD * 4, stream);
  hipMemsetAsync(cnt, 0, (size_t)G * S * 4, stream);

  const int nel = ntot * D;
  k_embed<<<(nel + 255) / 256, 256, 0, stream>>>(sp, emb, x, ntot, S, D);

  for (int l = 0; l < L; ++l) {
    k_init_h<<<(nel + 255) / 256, 256, 0, stream>>>(x, h, eps, l, nel);
    int nsc = nDir * (D / 4);
    k_scatter<<<(nsc + 255) / 256, 256, 0, stream>>>(ei, x, h, nDir, D);
    k_mlp_ln<<<(ntot + 127) / 128, 256, 0, stream>>>(
        h, x, W1 + (size_t)l * D * D, b1 + (size_t)l * D,
        W2 + (size_t)l * D * D, b2 + (size_t)l * D,
        lng + (size_t)l * D, lnb + (size_t)l * D, ntot);
  }

  k_pool_scatter<<<(nel + 255) / 256, 256, 0, stream>>>(x, sp, leaf, batch, P, cnt,
                                                        ntot, S, G, D);
  int npd = S * GD;
  k_pool_div<<<(npd + 255) / 256, 256, 0, stream>>>(P, cnt, npd, G, GD, S);

  int ncc = nE * G;
  k_ccnt<<<(ncc + 255) / 256, 256, 0, stream>>>(clade, cnt, ccn, nE, G, S);

  dim3 ggrid((nE + 15) / 16, (GD + 511) / 512);
  size_t asz = (size_t)(S / 2) * 16 * sizeof(v2f);
  k_clade_gemm<<<ggrid, 256, asz, stream>>>(clade, P, csum, nE, S, GD);

  int fthreads = 64;
  size_t fsm = (size_t)(2 * G + fthreads) * sizeof(float);
  k_finalize<<<nE, fthreads, fsm, stream>>>(csum, ccn, out, G, D);
}